// Net_55559696941167
// MI455X (gfx1250) — compile-verified
//
#include <hip/hip_runtime.h>
#include <hip/hip_bf16.h>
#include <math.h>

// Problem constants (match the reference)
#define KDIM 128
#define EXER 16384
#define NENT (KDIM + EXER)      // 16512
#define BATCH 2048
#define HLEN 64
#define NEDGE 262144

typedef __attribute__((ext_vector_type(2))) float v2f;
typedef __attribute__((ext_vector_type(8))) float v8f;

__device__ __forceinline__ float sigmoidf_(float x) {
    return 1.0f / (1.0f + expf(-x));
}

// ---------------------------------------------------------------- utilities
__global__ void fill_zero_kernel(float* __restrict__ p, size_t n) {
    size_t i = (size_t)blockIdx.x * blockDim.x + threadIdx.x;
    size_t stride = (size_t)gridDim.x * blockDim.x;
    for (; i < n; i += stride) p[i] = 0.0f;
}

// ------------------------------------------------------- degree + edge norm
__global__ void count_deg_kernel(const int* __restrict__ src,
                                 const int* __restrict__ dst,
                                 float* __restrict__ deg_out,
                                 float* __restrict__ deg_in, int ne) {
    int e = blockIdx.x * blockDim.x + threadIdx.x;
    if (e >= ne) return;
    atomicAdd(&deg_out[src[e]], 1.0f);
    atomicAdd(&deg_in[dst[e]], 1.0f);
}

__global__ void edge_norm_kernel(const int* __restrict__ src,
                                 const int* __restrict__ dst,
                                 const float* __restrict__ deg_out,
                                 const float* __restrict__ deg_in,
                                 float* __restrict__ norm, int ne) {
    int e = blockIdx.x * blockDim.x + threadIdx.x;
    if (e >= ne) return;
    float o = fmaxf(deg_out[src[e]], 1.0f);
    float i = fmaxf(deg_in[dst[e]], 1.0f);
    norm[e] = rsqrtf(o * i);
}

// ------------------------------------------------------ edge message scatter
// one wave32 per edge; each lane handles 4 consecutive columns (float4 load)
__global__ void scatter_msgs_kernel(const float* __restrict__ h,
                                    const int* __restrict__ src,
                                    const int* __restrict__ dst,
                                    const float* __restrict__ norm,
                                    float* __restrict__ agg, int ne) {
    int t = blockIdx.x * blockDim.x + threadIdx.x;
    int e = t >> 5;
    if (e >= ne) return;
    int lane = t & 31;
    int s = src[e];
    int d = dst[e];
    float w = norm[e];
    const float4 v = *(const float4*)(h + (size_t)s * KDIM + lane * 4);
    float* o = agg + (size_t)d * KDIM + lane * 4;
    atomicAdd(o + 0, v.x * w);
    atomicAdd(o + 1, v.y * w);
    atomicAdd(o + 2, v.z * w);
    atomicAdd(o + 3, v.w * w);
}

// ------------------------------------------------------------- WMMA GEMM
// out[N x K] = sum_i agg_i[N x K] @ W_i[K x K] + sum_i b_i
// fp32 WMMA 16x16x4; one wave per 16x16 C tile, 96 WMMAs along k=0..383.
__global__ __launch_bounds__(128) void gemm_gcn_kernel(
        const float* __restrict__ agg,   // 3 * NENT * KDIM
        const float* __restrict__ Wl,    // 3 * KDIM * KDIM (this layer)
        const float* __restrict__ bl,    // 3 * KDIM       (this layer)
        float* __restrict__ out)         // NENT * KDIM
{
    const int wavesPerBlock = 128 >> 5;                       // 4
    int wave = blockIdx.x * wavesPerBlock + (threadIdx.x >> 5);
    int lane = threadIdx.x & 31;
    const int ntN = KDIM / 16;                                // 8
    int mt = wave / ntN;
    int nt = wave % ntN;
    int m0 = mt * 16;
    int n0 = nt * 16;
    int half = lane >> 4;   // 0: K offsets {0,1}; 1: K offsets {2,3}
    int l16 = lane & 15;    // A: row M; B/C: column N

    v8f c = {0.f, 0.f, 0.f, 0.f, 0.f, 0.f, 0.f, 0.f};

    for (int i = 0; i < 3; ++i) {
        const float* A  = agg + (size_t)i * NENT * KDIM + (size_t)(m0 + l16) * KDIM;
        const float* Wi = Wl  + (size_t)i * KDIM * KDIM;
        for (int k = 0; k < KDIM; k += 4) {
            int ka = k + 2 * half;
            // A tile 16x4: lane row = l16, two consecutive k elements
            v2f a = *(const v2f*)(A + ka);
            // B tile 4x16: VGPR0 -> row ka, VGPR1 -> row ka+1, col = n0+l16
            v2f b;
            b.x = Wi[(size_t)ka * KDIM + n0 + l16];
            b.y = Wi[(size_t)(ka + 1) * KDIM + n0 + l16];
            c = __builtin_amdgcn_wmma_f32_16x16x4_f32(
                    false, a, false, b, (short)0, c, false, false);
        }
    }

    int col = n0 + l16;
    float bias = bl[col] + bl[KDIM + col] + bl[2 * KDIM + col];
    // C/D layout: VGPR r -> row m0 + r + 8*half, col = n0 + l16
    float* orow = out + (size_t)(m0 + 8 * half) * KDIM + col;
#pragma unroll
    for (int r = 0; r < 8; ++r)
        orow[(size_t)r * KDIM] = c[r] + bias;
}

// --------------------------------------------- concept projections (K x K)
// cW_stu[j] = sum_k concept[j][k]*Wstu[K+k];   cW_exer likewise.
// concept = 0.25*(entity + e1 + e2 + e3) rows [0..K)
__global__ void concept_kernel(const float* __restrict__ ent,
                               const float* __restrict__ e1,
                               const float* __restrict__ e2,
                               const float* __restrict__ e3,
                               const float* __restrict__ Wstu,
                               const float* __restrict__ Wexer,
                               float* __restrict__ cW_stu,
                               float* __restrict__ cW_exer) {
    int j = threadIdx.x;   // 0..127, single block
    size_t base = (size_t)j * KDIM;
    float s1 = 0.f, s2 = 0.f;
    for (int k = 0; k < KDIM; ++k) {
        float cv = 0.25f * (ent[base + k] + e1[base + k] + e2[base + k] + e3[base + k]);
        s1 += cv * Wstu[KDIM + k];
        s2 += cv * Wexer[KDIM + k];
    }
    cW_stu[j] = s1;
    cW_exer[j] = s2;
}

// --------------------------------------- masked history mean  (B x K output)
// stu_exe row r = 0.5*(e2 + e3)[K + r]
__global__ void hist_mean_kernel(const float* __restrict__ e2,
                                 const float* __restrict__ e3,
                                 const int* __restrict__ history,
                                 const int* __restrict__ history_len,
                                 float* __restrict__ stu_emb) {
    int b = blockIdx.x;
    int k = threadIdx.x;
    int len = history_len[b];
    float acc = 0.f;
    for (int h = 0; h < len; ++h) {
        int id = history[b * HLEN + h];
        size_t row = (size_t)(KDIM + id) * KDIM + k;
        acc += 0.5f * (e2[row] + e3[row]);
    }
    stu_emb[(size_t)b * KDIM + k] = acc / (float)len;
}

// -------------------------------------------------------------- final head
__global__ void final_kernel(const float* __restrict__ ent,
                             const float* __restrict__ e1,
                             const float* __restrict__ e2,
                             const float* __restrict__ e3,
                             const float* __restrict__ stu_emb,
                             const float* __restrict__ cW_stu,
                             const float* __restrict__ cW_exer,
                             const int* __restrict__ exer_id,
                             const float* __restrict__ disc_table,
                             const float* __restrict__ Wstu,
                             const float* __restrict__ bstu,
                             const float* __restrict__ Wexer,
                             const float* __restrict__ bexer,
                             const float* __restrict__ W3,
                             const float* __restrict__ b3,
                             const float* __restrict__ kn_emb,
                             float* __restrict__ outp) {
    __shared__ float r1[KDIM];
    __shared__ float r2[KDIM];
    int b = blockIdx.x;
    int k = threadIdx.x;
    int eid = exer_id[b];
    size_t row = (size_t)(KDIM + eid) * KDIM + k;
    float ex = 0.25f * (ent[row] + e1[row] + e2[row] + e3[row]);   // exer_emb[k]
    float se = stu_emb[(size_t)b * KDIM + k];

    r1[k] = se * Wstu[k];
    r2[k] = ex * Wexer[k];
    __syncthreads();
    for (int off = KDIM / 2; off > 0; off >>= 1) {
        if (k < off) { r1[k] += r1[k + off]; r2[k] += r2[k + off]; }
        __syncthreads();
    }
    float s_stu = r1[0];
    float s_ex  = r2[0];
    __syncthreads();

    float prof = sigmoidf_(s_stu + cW_stu[k] + bstu[0]);
    float kd   = sigmoidf_(s_ex  + cW_exer[k] + bexer[0]);
    float disc = sigmoidf_(disc_table[eid]) * 10.0f;
    float x = disc * (prof - kd) * kn_emb[(size_t)b * KDIM + k];

    r1[k] = x * W3[k];
    __syncthreads();
    for (int off = KDIM / 2; off > 0; off >>= 1) {
        if (k < off) r1[k] += r1[k + off];
        __syncthreads();
    }
    if (k == 0) outp[b] = sigmoidf_(r1[0] + b3[0]);
}

// ================================================================= launcher
extern "C" void kernel_launch(void* const* d_in, const int* in_sizes, int n_in,
                              void* d_out, int out_size, void* d_ws, size_t ws_size,
                              hipStream_t stream) {
    const float* entity     = (const float*)d_in[0];
    const float* gcn_W      = (const float*)d_in[1];
    const float* gcn_b      = (const float*)d_in[2];
    const float* disc_table = (const float*)d_in[3];
    const float* Wstu       = (const float*)d_in[4];
    const float* bstu       = (const float*)d_in[5];
    const float* Wexer      = (const float*)d_in[6];
    const float* bexer      = (const float*)d_in[7];
    const float* W3         = (const float*)d_in[8];
    const float* b3         = (const float*)d_in[9];
    const float* kn_emb     = (const float*)d_in[10];
    // d_in[11] stu_id, d_in[13] input_knowedge_ids: unused by the reference
    const int* exer_id      = (const int*)d_in[12];
    const int* history      = (const int*)d_in[14];
    const int* history_len  = (const int*)d_in[15];
    const int* srcs[3] = { (const int*)d_in[16], (const int*)d_in[18], (const int*)d_in[20] };
    const int* dsts[3] = { (const int*)d_in[17], (const int*)d_in[19], (const int*)d_in[21] };
    float* outp = (float*)d_out;

    const size_t NK = (size_t)NENT * KDIM;

    // workspace layout (floats)
    float* f = (float*)d_ws;
    float* norm[3];
    norm[0] = f;                 // E
    norm[1] = norm[0] + NEDGE;   // E
    norm[2] = norm[1] + NEDGE;   // E
    float* deg_out = norm[2] + NEDGE;   // N
    float* deg_in  = deg_out + NENT;    // N
    float* e1 = deg_in + NENT;          // N*K
    float* e2 = e1 + NK;                // N*K
    float* e3 = e2 + NK;                // N*K
    float* agg = e3 + NK;               // 3*N*K
    float* cW_stu = agg + 3 * NK;       // K
    float* cW_exer = cW_stu + KDIM;     // K
    float* stu_emb = cW_exer + KDIM;    // B*K
    (void)ws_size; (void)n_in; (void)in_sizes; (void)out_size;

    // 1) per-graph edge norms (degrees are layer-invariant)
    for (int g = 0; g < 3; ++g) {
        fill_zero_kernel<<<64, 256, 0, stream>>>(deg_out, (size_t)2 * NENT);
        count_deg_kernel<<<(NEDGE + 255) / 256, 256, 0, stream>>>(
            srcs[g], dsts[g], deg_out, deg_in, NEDGE);
        edge_norm_kernel<<<(NEDGE + 255) / 256, 256, 0, stream>>>(
            srcs[g], dsts[g], deg_out, deg_in, norm[g], NEDGE);
    }

    // 2) three GCN layers (layer 3 of the reference is dead code)
    const float* hin[4] = { entity, e1, e2, e3 };
    float* hout[3] = { e1, e2, e3 };
    const int scatterBlocks = (NEDGE * 32 + 255) / 256;
    const int gemmWaves = (NENT / 16) * (KDIM / 16);   // 8256
    const int gemmBlocks = gemmWaves / 4;              // 4 waves per block

    for (int l = 0; l < 3; ++l) {
        fill_zero_kernel<<<512, 256, 0, stream>>>(agg, 3 * NK);
        for (int g = 0; g < 3; ++g) {
            scatter_msgs_kernel<<<scatterBlocks, 256, 0, stream>>>(
                hin[l], srcs[g], dsts[g], norm[g], agg + (size_t)g * NK, NEDGE);
        }
        gemm_gcn_kernel<<<gemmBlocks, 128, 0, stream>>>(
            agg,
            gcn_W + (size_t)l * 3 * KDIM * KDIM,
            gcn_b + (size_t)l * 3 * KDIM,
            hout[l]);
    }

    // 3) head
    concept_kernel<<<1, KDIM, 0, stream>>>(entity, e1, e2, e3, Wstu, Wexer,
                                           cW_stu, cW_exer);
    hist_mean_kernel<<<BATCH, KDIM, 0, stream>>>(e2, e3, history, history_len,
                                                 stu_emb);
    final_kernel<<<BATCH, KDIM, 0, stream>>>(entity, e1, e2, e3, stu_emb,
                                             cW_stu, cW_exer, exer_id,
                                             disc_table, Wstu, bstu, Wexer,
                                             bexer, W3, b3, kn_emb, outp);
}